// GAT_StructuralAttention_39608188404041
// MI455X (gfx1250) — compile-verified
//
#include <hip/hip_runtime.h>
#include <math.h>

// ---------------------------------------------------------------------------
// GAT structural attention, MI455X (gfx1250, wave32, WMMA)
// N=20000, E=320000, IN=H*D=128, H=8, D=16, L=2, OUT=2
// ---------------------------------------------------------------------------

constexpr int kN   = 20000;
constexpr int kE   = 320000;
constexpr int kIN  = 128;
constexpr int kH   = 8;
constexpr int kD   = 16;
constexpr int kHD  = 128;   // H*D == IN (residual always taken)
constexpr int kL   = 2;
constexpr int kOUT = 2;
constexpr float kLnEps = 1e-5f;

typedef __attribute__((ext_vector_type(2))) float v2f;
typedef __attribute__((ext_vector_type(8))) float v8f;

__device__ __forceinline__ float eluf(float x) {
    return x > 0.0f ? x : (expf(x) - 1.0f);
}

// ---------------------------------------------------------------------------
// GEMM: H1 = X*W1 + b1, H2 = X*W2 + b2, HP = H1.*H2    (M=20000, K=N=128)
// grid.x = 1250 M-tiles of 16 rows; 8 waves/block, wave w -> cols [16w,16w+16)
// Uses V_WMMA_F32_16X16X4_F32 (exact fp32), K-loop of 32 steps, 2 WMMA each.
// A 16x4 f32 layout: lanes 0-15 hold K={k0,k0+1}, lanes 16-31 K={k0+2,k0+3},
// row M = lane%16.  B 4x16 mirrors with N = lane%16.
// C/D: VGPR j -> M = j + 8*(lane>=16), N = lane%16.
// ---------------------------------------------------------------------------
__global__ __launch_bounds__(256) void gemm_h1h2_kernel(
    const float* __restrict__ X,
    const float* __restrict__ W1, const float* __restrict__ B1,
    const float* __restrict__ W2, const float* __restrict__ B2,
    float* __restrict__ H1, float* __restrict__ H2, float* __restrict__ HP)
{
    const int lane  = threadIdx.x & 31;
    const int wave  = threadIdx.x >> 5;
    const int mbase = blockIdx.x * 16;
    const int n0    = wave * 16;
    const int r     = lane & 15;
    const int half  = lane >> 4;          // 0: K pair {0,1}, 1: K pair {2,3}

    const float* xrow = X + (size_t)(mbase + r) * kIN;

    v8f acc1 = {};
    v8f acc2 = {};

#pragma unroll
    for (int k0 = 0; k0 < kIN; k0 += 4) {
        const int ka = k0 + half * 2;
        v2f a = *(const v2f*)(xrow + ka);              // 8B-aligned (ka even)
        v2f bA, bB;
        bA.x = W1[(size_t)ka       * kHD + n0 + r];
        bA.y = W1[(size_t)(ka + 1) * kHD + n0 + r];
        bB.x = W2[(size_t)ka       * kHD + n0 + r];
        bB.y = W2[(size_t)(ka + 1) * kHD + n0 + r];
        acc1 = __builtin_amdgcn_wmma_f32_16x16x4_f32(
            false, a, false, bA, (short)0, acc1, false, false);
        acc2 = __builtin_amdgcn_wmma_f32_16x16x4_f32(
            false, a, false, bB, (short)0, acc2, false, false);
    }

    const int n = n0 + r;
    const float bb1 = B1[n];
    const float bb2 = B2[n];

#pragma unroll
    for (int j = 0; j < 8; ++j) {
        const int m = mbase + j + half * 8;
        const float h1v = acc1[j] + bb1;
        const float h2v = acc2[j] + bb2;
        const size_t idx = (size_t)m * kHD + n;
        H1[idx] = h1v;
        H2[idx] = h2v;
        HP[idx] = h1v * h2v;
    }
}

// ---------------------------------------------------------------------------
// Edge kernel: one wave32 per edge.  Lane l owns elements [4l, 4l+4) of the
// 128-wide feature row; lane group-of-4 == one head (h = l/4, a-flat = 4l).
//   t      = elu(h1[src] + h2[dst] + hp[src]*hp[dst])
//   logit  = <a_h, t_h> / sqrt(D)
//   alpha  = softmax over the 8 HEADS (per edge)
//   agg[dst] += alpha_h * h1[src]           (f32 global atomics, L2-resident)
// ---------------------------------------------------------------------------
__global__ __launch_bounds__(256) void edge_attn_kernel(
    const float* __restrict__ H1, const float* __restrict__ H2,
    const float* __restrict__ HP,
    const int* __restrict__ src, const int* __restrict__ dst,
    const float* __restrict__ avec, float* __restrict__ AGG)
{
    const int lane = threadIdx.x & 31;
    const int e    = blockIdx.x * 8 + (threadIdx.x >> 5);
    if (e >= kE) return;

    const int s = src[e];
    const int d = dst[e];

    const float4 h1s = *(const float4*)(H1 + (size_t)s * kHD + lane * 4);
    const float4 h2d = *(const float4*)(H2 + (size_t)d * kHD + lane * 4);
    const float4 hps = *(const float4*)(HP + (size_t)s * kHD + lane * 4);
    const float4 hpd = *(const float4*)(HP + (size_t)d * kHD + lane * 4);
    const float4 av  = *(const float4*)(avec + lane * 4);

    float part;
    {
        const float t0 = eluf(h1s.x + h2d.x + hps.x * hpd.x);
        const float t1 = eluf(h1s.y + h2d.y + hps.y * hpd.y);
        const float t2 = eluf(h1s.z + h2d.z + hps.z * hpd.z);
        const float t3 = eluf(h1s.w + h2d.w + hps.w * hpd.w);
        part = av.x * t0 + av.y * t1 + av.z * t2 + av.w * t3;
    }
    part *= 0.25f;                       // 1/sqrt(D), D = 16

    // all-reduce within the head's group-of-4 lanes -> per-head logit
    part += __shfl_xor(part, 1, 32);
    part += __shfl_xor(part, 2, 32);

    // softmax across the 8 heads: xor strides {4,8,16} touch each head once
    float mx = part;
    mx = fmaxf(mx, __shfl_xor(mx, 4, 32));
    mx = fmaxf(mx, __shfl_xor(mx, 8, 32));
    mx = fmaxf(mx, __shfl_xor(mx, 16, 32));

    const float ex = expf(part - mx);
    float sm = ex;
    sm += __shfl_xor(sm, 4, 32);
    sm += __shfl_xor(sm, 8, 32);
    sm += __shfl_xor(sm, 16, 32);

    const float alpha = ex / sm;

    float* aggrow = AGG + (size_t)d * kHD + lane * 4;
    atomicAdd(aggrow + 0, alpha * h1s.x);
    atomicAdd(aggrow + 1, alpha * h1s.y);
    atomicAdd(aggrow + 2, alpha * h1s.z);
    atomicAdd(aggrow + 3, alpha * h1s.w);
}

// ---------------------------------------------------------------------------
// LayerNorm(128) + ELU + residual, one wave per node (4 elems/lane).
// ---------------------------------------------------------------------------
__global__ __launch_bounds__(256) void ln_elu_res_kernel(
    const float* __restrict__ AGG,
    const float* __restrict__ G, const float* __restrict__ B,
    const float* __restrict__ Xin, float* __restrict__ Xout)
{
    const int lane = threadIdx.x & 31;
    const int node = blockIdx.x * 8 + (threadIdx.x >> 5);
    if (node >= kN) return;

    const float4 v = *(const float4*)(AGG + (size_t)node * kHD + lane * 4);

    float s = v.x + v.y + v.z + v.w;
#pragma unroll
    for (int m = 1; m <= 16; m <<= 1) s += __shfl_xor(s, m, 32);
    const float mu = s * (1.0f / 128.0f);

    const float dx0 = v.x - mu, dx1 = v.y - mu, dx2 = v.z - mu, dx3 = v.w - mu;
    float q = dx0 * dx0 + dx1 * dx1 + dx2 * dx2 + dx3 * dx3;
#pragma unroll
    for (int m = 1; m <= 16; m <<= 1) q += __shfl_xor(q, m, 32);
    const float inv = rsqrtf(q * (1.0f / 128.0f) + kLnEps);

    const float4 g  = *(const float4*)(G + lane * 4);
    const float4 b  = *(const float4*)(B + lane * 4);
    const float4 xr = *(const float4*)(Xin + (size_t)node * kHD + lane * 4);

    float4 o;
    o.x = eluf(dx0 * inv * g.x + b.x) + xr.x;
    o.y = eluf(dx1 * inv * g.y + b.y) + xr.y;
    o.z = eluf(dx2 * inv * g.z + b.z) + xr.z;
    o.w = eluf(dx3 * inv * g.w + b.w) + xr.w;

    *(float4*)(Xout + (size_t)node * kHD + lane * 4) = o;
}

// ---------------------------------------------------------------------------
// Final projection: out[n,:] = x[n,:] @ Wo[128,2] + bo.  One thread per node.
// ---------------------------------------------------------------------------
__global__ __launch_bounds__(256) void out_proj_kernel(
    const float* __restrict__ X, const float* __restrict__ Wo,
    const float* __restrict__ bo, float* __restrict__ O)
{
    const int n = blockIdx.x * blockDim.x + threadIdx.x;
    if (n >= kN) return;
    float a0 = bo[0];
    float a1 = bo[1];
    const float* xr = X + (size_t)n * kHD;
#pragma unroll 8
    for (int k = 0; k < kHD; ++k) {
        const float xv = xr[k];
        a0 += xv * Wo[2 * k + 0];
        a1 += xv * Wo[2 * k + 1];
    }
    O[2 * n + 0] = a0;
    O[2 * n + 1] = a1;
}

// ---------------------------------------------------------------------------
extern "C" void kernel_launch(void* const* d_in, const int* in_sizes, int n_in,
                              void* d_out, int out_size, void* d_ws, size_t ws_size,
                              hipStream_t stream)
{
    (void)in_sizes; (void)n_in; (void)out_size; (void)ws_size;

    const float* x    = (const float*)d_in[0];
    const int*   ei   = (const int*)  d_in[1];   // [2,E]: src row then dst row
    const float* W1   = (const float*)d_in[2];   // [L,128,128]
    const float* b1   = (const float*)d_in[3];   // [L,128]
    const float* W2   = (const float*)d_in[4];
    const float* b2   = (const float*)d_in[5];
    const float* a    = (const float*)d_in[6];   // [L,8,16]
    const float* ln_g = (const float*)d_in[7];
    const float* ln_b = (const float*)d_in[8];
    const float* Wo   = (const float*)d_in[9];   // [128,2]
    const float* bo   = (const float*)d_in[10];
    float*       out  = (float*)d_out;

    const size_t NHD = (size_t)kN * kHD;         // 2,560,000 floats
    float* h1   = (float*)d_ws;
    float* h2   = h1  + NHD;
    float* hp   = h2  + NHD;
    float* agg  = hp  + NHD;
    float* xcur = agg + NHD;

    const int* srcIdx = ei;
    const int* dstIdx = ei + kE;

    for (int layer = 0; layer < kL; ++layer) {
        const float* xin = (layer == 0) ? x : xcur;

        gemm_h1h2_kernel<<<kN / 16, 256, 0, stream>>>(
            xin,
            W1 + (size_t)layer * kIN * kHD, b1 + (size_t)layer * kHD,
            W2 + (size_t)layer * kIN * kHD, b2 + (size_t)layer * kHD,
            h1, h2, hp);

        hipMemsetAsync(agg, 0, NHD * sizeof(float), stream);

        edge_attn_kernel<<<kE / 8, 256, 0, stream>>>(
            h1, h2, hp, srcIdx, dstIdx, a + (size_t)layer * kH * kD, agg);

        ln_elu_res_kernel<<<kN / 8, 256, 0, stream>>>(
            agg, ln_g + (size_t)layer * kHD, ln_b + (size_t)layer * kHD,
            xin, xcur);
    }

    out_proj_kernel<<<(kN + 255) / 256, 256, 0, stream>>>(xcur, Wo, bo, out);
}